// Wavelet_42898133353246
// MI455X (gfx1250) — compile-verified
//
#include <hip/hip_runtime.h>
#include <stdint.h>

#define IS2        0.70710678118654752440f   // 1/sqrt(2)
#define SQRT_2LOGL 3.7232978f                // sqrt(2*ln(1024))
#define INV_MAD    1.4825797f                // 1/0.6745
#define INV_LOG2_3 0.63092975f               // 1/log2(3)

__device__ __forceinline__ float soft_thr(float c, float t) {
    float a = __builtin_fabsf(c);
    return (a >= t) ? __builtin_copysignf(a - t, c) : 0.0f;
}

// One wave32 per row of 1024 fp32. 8 rows per 256-thread block.
__global__ __launch_bounds__(256) void haar_denoise_rows(
    const float* __restrict__ x, float* __restrict__ y, int n_rows)
{
    __shared__ float4 sbuf[8 * 256];          // 4 KB staging per wave, 32 KB/block
    const int lane = threadIdx.x & 31;
    const int wv   = threadIdx.x >> 5;
    const int row  = blockIdx.x * 8 + wv;
    if (row >= n_rows) return;                // wave-uniform

    const float4* gin  = (const float4*)x + (size_t)row * 256;
    float4*       gout = (float4*)y       + (size_t)row * 256;
    float4*       sw   = sbuf + wv * 256;

    // ---- coalesced async load: global chunk g=lane+32*i -> LDS slot (g>>3)+32*(g&7)
    // (so per-lane contiguous reads below are conflict-free)  [ASYNCcnt path]
    // Streaming data, no reuse: non-temporal hint protects the 192MB L2.
    #pragma unroll
    for (int i = 0; i < 8; ++i) {
        int g = lane + 32 * i;
        unsigned lds_addr = (unsigned)(uintptr_t)&sw[(g >> 3) + 32 * (g & 7)];
        unsigned long long gaddr = (unsigned long long)(uintptr_t)(gin + g);
        asm volatile("global_load_async_to_lds_b128 %0, %1, off th:TH_LOAD_NT"
                     :: "v"(lds_addr), "v"(gaddr) : "memory");
    }
    asm volatile("s_wait_asynccnt 0" ::: "memory");

    // ---- each lane pulls its contiguous 32 elements (chunk 8*lane+i lives at slot lane+32*i)
    float xr[32];
    #pragma unroll
    for (int i = 0; i < 8; ++i) {
        float4 v = sw[lane + 32 * i];
        xr[4*i+0] = v.x; xr[4*i+1] = v.y; xr[4*i+2] = v.z; xr[4*i+3] = v.w;
    }

    // ---- 3-level Haar DWT, fully lane-local
    float ca1[16], cd1[16];
    #pragma unroll
    for (int i = 0; i < 16; ++i) {
        ca1[i] = (xr[2*i] + xr[2*i+1]) * IS2;
        cd1[i] = (xr[2*i] - xr[2*i+1]) * IS2;
    }
    float ca2[8], cd2[8];
    #pragma unroll
    for (int i = 0; i < 8; ++i) {
        ca2[i] = (ca1[2*i] + ca1[2*i+1]) * IS2;
        cd2[i] = (ca1[2*i] - ca1[2*i+1]) * IS2;
    }
    float ca3[4], cd3[4];
    #pragma unroll
    for (int i = 0; i < 4; ++i) {
        ca3[i] = (ca2[2*i] + ca2[2*i+1]) * IS2;
        cd3[i] = (ca2[2*i] - ca2[2*i+1]) * IS2;
    }

    // ---- exact median of |cd1| over the 512 row values:
    // |v| bit patterns are order-isomorphic uints; dual-rank (255,256) radix bisection.
    unsigned au[16];
    #pragma unroll
    for (int i = 0; i < 16; ++i) au[i] = __float_as_uint(cd1[i]) & 0x7fffffffu;

    unsigned p0 = 0u, p1 = 0u;
    for (int b = 30; b >= 0; --b) {
        unsigned t0 = p0 | (1u << b);
        unsigned t1 = p1 | (1u << b);
        int c = 0;
        #pragma unroll
        for (int i = 0; i < 16; ++i) {
            c += (au[i] < t0) ? 1 : 0;          // count for rank 255 (low 16 bits)
            c += (au[i] < t1) ? 0x10000 : 0;    // count for rank 256 (high 16 bits)
        }
        #pragma unroll
        for (int s = 16; s >= 1; s >>= 1) c += __shfl_xor(c, s, 32);
        if ((c & 0xffff) <= 255) p0 = t0;       // wave-uniform branches
        if ((c >> 16)    <= 256) p1 = t1;
    }
    float median = 0.5f * (__uint_as_float(p0) + __uint_as_float(p1));
    float lamda  = (median * INV_MAD) * SQRT_2LOGL;
    float thr1 = lamda;                 // / log2(2)
    float thr2 = lamda * INV_LOG2_3;    // / log2(3)
    float thr3 = lamda * 0.5f;          // / log2(4)

    // ---- soft-threshold detail bands
    #pragma unroll
    for (int i = 0; i < 16; ++i) cd1[i] = soft_thr(cd1[i], thr1);
    #pragma unroll
    for (int i = 0; i < 8;  ++i) cd2[i] = soft_thr(cd2[i], thr2);
    #pragma unroll
    for (int i = 0; i < 4;  ++i) cd3[i] = soft_thr(cd3[i], thr3);

    // ---- 3-level inverse DWT, lane-local
    float ca2r[8];
    #pragma unroll
    for (int i = 0; i < 4; ++i) {
        ca2r[2*i]   = (ca3[i] + cd3[i]) * IS2;
        ca2r[2*i+1] = (ca3[i] - cd3[i]) * IS2;
    }
    float ca1r[16];
    #pragma unroll
    for (int i = 0; i < 8; ++i) {
        ca1r[2*i]   = (ca2r[i] + cd2[i]) * IS2;
        ca1r[2*i+1] = (ca2r[i] - cd2[i]) * IS2;
    }

    // ---- final level -> stage into LDS (conflict-free contiguous slots)
    #pragma unroll
    for (int i = 0; i < 8; ++i) {
        float a0 = ca1r[2*i],   d0 = cd1[2*i];
        float a1 = ca1r[2*i+1], d1 = cd1[2*i+1];
        float4 o;
        o.x = (a0 + d0) * IS2;  o.y = (a0 - d0) * IS2;
        o.z = (a1 + d1) * IS2;  o.w = (a1 - d1) * IS2;
        sw[lane + 32 * i] = o;                  // chunk 8*lane+i at slot lane+32*i
    }
    asm volatile("s_wait_dscnt 0" ::: "memory");  // LDS stores visible to async engine

    // ---- coalesced async store: LDS slot (g>>3)+32*(g&7) -> global chunk g=lane+32*i
    // Output is never re-read: stream it past the caches.
    #pragma unroll
    for (int i = 0; i < 8; ++i) {
        int g = lane + 32 * i;
        unsigned lds_addr = (unsigned)(uintptr_t)&sw[(g >> 3) + 32 * (g & 7)];
        unsigned long long gaddr = (unsigned long long)(uintptr_t)(gout + g);
        asm volatile("global_store_async_from_lds_b128 %0, %1, off th:TH_STORE_NT"
                     :: "v"(gaddr), "v"(lds_addr) : "memory");
    }
    asm volatile("s_wait_asynccnt 0" ::: "memory");
}

extern "C" void kernel_launch(void* const* d_in, const int* in_sizes, int n_in,
                              void* d_out, int out_size, void* d_ws, size_t ws_size,
                              hipStream_t stream) {
    const float* x = (const float*)d_in[0];
    float* y = (float*)d_out;
    int n_rows = in_sizes[0] >> 10;             // rows of length 1024
    int blocks = (n_rows + 7) / 8;              // 8 row-waves per 256-thread block
    haar_denoise_rows<<<blocks, 256, 0, stream>>>(x, y, n_rows);
}